// SemiConv2d_10230612099335
// MI455X (gfx1250) — compile-verified
//
#include <hip/hip_runtime.h>
#include <stdint.h>

// Problem constants (fixed by the reference harness)
#define BATCH   4
#define CIN     32
#define COUT    32
#define HH      80
#define WW      80
#define TAPS    25          // 5x5
#define TILE    16          // spatial tile (outputs) per block, 16x16
#define HALO    20          // TILE + 4 (pad=2 each side)
#define CO_SPLIT 8          // output channels per block
#define WS_STRIDE 32        // taps padded 25 -> 32 floats for 16B-aligned ds_load_b128

#define XS_ELEMS (CIN * HALO * HALO)              // 12800 floats
#define WS_ELEMS (CO_SPLIT * CIN * WS_STRIDE)     // 8192 floats
#define XS_BYTES (XS_ELEMS * 4)                   // 51200
#define LDS_BYTES (XS_BYTES + WS_ELEMS * 4)       // 83968 (3 blocks / 320KB WGP)

typedef float v2f __attribute__((ext_vector_type(2)));
typedef float v4f __attribute__((ext_vector_type(4)));

__device__ __forceinline__ float max3f(float a, float b, float c) {
    // max(max(a,b),c) -> v_max3_num_f32 on gfx1250
    return __builtin_fmaxf(__builtin_fmaxf(a, b), c);
}

// CDNA5 async global->LDS copy (per-lane scatter), tracked by ASYNCcnt.
__device__ __forceinline__ void async_ld_b32(uint32_t lds_off, const float* g) {
    uint64_t ga = (uint64_t)(uintptr_t)g;
    asm volatile("global_load_async_to_lds_b32 %0, %1, off"
                 :: "v"(lds_off), "v"(ga) : "memory");
}

__global__ __launch_bounds__(256)
void semiconv2d_maxplus_kernel(const float* __restrict__ x,
                               const float* __restrict__ w,
                               const float* __restrict__ bias,
                               float* __restrict__ out)
{
    extern __shared__ float smem[];
    float* xs = smem;                  // [CIN][HALO][HALO], -inf outside image
    float* ws = smem + XS_ELEMS;       // [CO_SPLIT][CIN][WS_STRIDE]

    const int tid = threadIdx.x;
    int bx = blockIdx.x;               // 400 blocks: 25 tiles x 4 co-slices x 4 batches
    const int tile_x = bx % 5;  bx /= 5;
    const int tile_y = bx % 5;  bx /= 5;
    const int cog    = bx & 3;  bx >>= 2;
    const int b      = bx;             // 0..3
    const int co0    = cog * CO_SPLIT;

    const uint32_t lds_base = (uint32_t)(uintptr_t)smem;  // flat LDS addr low bits = LDS offset
    const float NEGINF = -__builtin_inff();

    // ---- 1) stage x tile: async-load valid cells, ds_store -inf to halo-out cells ----
    const int gy0 = tile_y * TILE - 2;
    const int gx0 = tile_x * TILE - 2;
    #pragma unroll 1
    for (int j = 0; j < XS_ELEMS / 256; ++j) {          // 50 iters
        int e  = tid + j * 256;
        int ci = e / (HALO * HALO);
        int r  = (e / HALO) % HALO;
        int c  = e % HALO;
        int gy = gy0 + r, gx = gx0 + c;
        if ((unsigned)gy < HH && (unsigned)gx < WW) {   // EXEC-masked async load
            const float* gp = x + (((size_t)(b * CIN + ci) * HH + gy) * WW + gx);
            async_ld_b32(lds_base + (uint32_t)e * 4u, gp);
        } else {
            xs[e] = NEGINF;                             // disjoint address: no hazard
        }
    }
    // weights: packed [co][ci][25] in memory -> strided [co][ci][32] in LDS (scatter)
    #pragma unroll 1
    for (int j = 0; j < (CO_SPLIT * CIN * TAPS) / 256; ++j) {   // 6400/256 = 25 iters
        int e   = tid + j * 256;
        int col = e / (CIN * TAPS);
        int rem = e % (CIN * TAPS);
        int ci  = rem / TAPS;
        int k   = rem % TAPS;
        const float* gp = w + (((size_t)(co0 + col) * CIN + ci) * TAPS + k);
        uint32_t loff = lds_base + XS_BYTES +
                        (uint32_t)((col * CIN + ci) * WS_STRIDE + k) * 4u;
        async_ld_b32(loff, gp);
    }
    asm volatile("s_wait_asynccnt 0" ::: "memory");
    __syncthreads();

    // ---- 2) compute: patch packed into f32 pairs once per ci, reused by 8 channels ----
    const int ty = tid >> 4;
    const int tx = tid & 15;

    float acc[CO_SPLIT];
    #pragma unroll
    for (int c = 0; c < CO_SPLIT; ++c) acc[c] = 0.0f;

    #pragma unroll 1
    for (int ci = 0; ci < CIN; ++ci) {
        float p[TAPS];
        const float* xp = xs + ci * (HALO * HALO) + ty * HALO + tx;
        #pragma unroll
        for (int m = 0; m < 5; ++m)
            #pragma unroll
            for (int n = 0; n < 5; ++n)
                p[m * 5 + n] = xp[m * HALO + n];

        v2f pp[12];                       // taps 0..23 as pairs -> v_pk_add_f32 operands
        #pragma unroll
        for (int i = 0; i < 12; ++i) { pp[i].x = p[2 * i]; pp[i].y = p[2 * i + 1]; }
        const float p24 = p[24];

        #pragma unroll
        for (int col = 0; col < CO_SPLIT; ++col) {
            const float* wp = ws + (col * CIN + ci) * WS_STRIDE;  // 16B aligned
            v4f wa = *(const v4f*)(wp +  0);
            v4f wb = *(const v4f*)(wp +  4);
            v4f wc = *(const v4f*)(wp +  8);
            v4f wd = *(const v4f*)(wp + 12);
            v4f we = *(const v4f*)(wp + 16);
            v4f wf = *(const v4f*)(wp + 20);
            float wg = wp[24];

            // 12x v_pk_add_f32 (2 taps each) + 1 scalar add
            v2f t0  = pp[0]  + wa.lo;     // s0  s1
            v2f t1  = pp[1]  + wa.hi;     // s2  s3
            v2f t2  = pp[2]  + wb.lo;     // s4  s5
            v2f t3  = pp[3]  + wb.hi;     // s6  s7
            v2f t4  = pp[4]  + wc.lo;     // s8  s9
            v2f t5  = pp[5]  + wc.hi;     // s10 s11
            v2f t6  = pp[6]  + wd.lo;     // s12 s13
            v2f t7  = pp[7]  + wd.hi;     // s14 s15
            v2f t8  = pp[8]  + we.lo;     // s16 s17
            v2f t9  = pp[9]  + we.hi;     // s18 s19
            v2f t10 = pp[10] + wf.lo;     // s20 s21
            v2f t11 = pp[11] + wf.hi;     // s22 s23
            float s24 = p24 + wg;

            // 12x v_max3_num_f32 tree over 25 values
            float g0 = max3f(t0.x,  t0.y,  t1.x);
            float g1 = max3f(t1.y,  t2.x,  t2.y);
            float g2 = max3f(t3.x,  t3.y,  t4.x);
            float g3 = max3f(t4.y,  t5.x,  t5.y);
            float g4 = max3f(t6.x,  t6.y,  t7.x);
            float g5 = max3f(t7.y,  t8.x,  t8.y);
            float g6 = max3f(t9.x,  t9.y,  t10.x);
            float g7 = max3f(t10.y, t11.x, t11.y);
            float h0 = max3f(g0, g1, g2);
            float h1 = max3f(g3, g4, g5);
            float h2 = max3f(g6, g7, s24);
            acc[col] += max3f(h0, h1, h2);
        }
    }

    // ---- 3) bias + coalesced store ----
    const int gh = tile_y * TILE + ty;
    const int gw = tile_x * TILE + tx;
    #pragma unroll
    for (int col = 0; col < CO_SPLIT; ++col) {
        float r = acc[col] + bias[co0 + col];
        out[(((size_t)(b * COUT) + co0 + col) * HH + gh) * WW + gw] = r;
    }
}

extern "C" void kernel_launch(void* const* d_in, const int* in_sizes, int n_in,
                              void* d_out, int out_size, void* d_ws, size_t ws_size,
                              hipStream_t stream) {
    (void)in_sizes; (void)n_in; (void)out_size; (void)d_ws; (void)ws_size;
    const float* x    = (const float*)d_in[0];
    const float* w    = (const float*)d_in[1];
    const float* bias = (const float*)d_in[2];
    float* out = (float*)d_out;

    // 25 spatial tiles x 4 co-slices x 4 batches = 400 blocks, 256 threads (8 wave32)
    semiconv2d_maxplus_kernel<<<400, 256, LDS_BYTES, stream>>>(x, w, bias, out);
}